// TorchMD_ET_dynamics_44641890075054
// MI455X (gfx1250) — compile-verified
//
#include <hip/hip_runtime.h>

#define HID    128
#define HEADS  8
#define HD     16
#define RBFN   50
#define KPAD   64
#define NNODES 20000
#define NEDGES 320000
#define ECHUNK 40000
#define NLAYERS 2
#define CUTF   10.0f

typedef float v2f __attribute__((ext_vector_type(2)));
typedef float v8f __attribute__((ext_vector_type(8)));

__device__ __forceinline__ float silu_f(float x) { return x / (1.0f + __expf(-x)); }

// ---------------------------------------------------------------------------
// Repack weight [krows,nc] (zero-padded to [kpad,nc]) into WMMA-B-native
// layout:  dst[((k/4)*nc + col)*4 + (k%4)] = src[k][col]
// so that each lane's B fragment (rows k+kh, k+kh+1 at column col) is one
// aligned 8-byte load, and a wave covers a contiguous 256B region per k-step.
// ---------------------------------------------------------------------------
__global__ void pack_weight_kernel(const float* __restrict__ src,
                                   float* __restrict__ dst,
                                   int krows, int kpad, int nc)
{
    const int p = blockIdx.x * blockDim.x + threadIdx.x;
    if (p >= kpad * nc) return;
    const int kk   = p & 3;
    const int rest = p >> 2;          // k4*nc + col
    const int col  = rest % nc;
    const int k4   = rest / nc;
    const int row  = (k4 << 2) + kk;
    dst[p] = (row < krows) ? src[row * nc + col] : 0.0f;
}

// ---------------------------------------------------------------------------
// WMMA f32 GEMM:  Y[M,NC] = act( X[M,K] @ W[K,NC] + bias (+ t*wext) )
// W must be pre-packed by pack_weight_kernel. One wave computes 16 rows;
// A-fragments cached in VGPRs across all N tiles. M % 16 == 0 at all call
// sites, and the M-guard is wave-uniform, so EXEC is all-ones around WMMA.
// ---------------------------------------------------------------------------
template<int K, int NC, int ACT, bool EXTRA>
__global__ __launch_bounds__(256) void gemm_wmma(
    const float* __restrict__ X, const float* __restrict__ Wp,
    const float* __restrict__ bias, float* __restrict__ Y,
    const float* __restrict__ tvec, const float* __restrict__ wext, int M)
{
    const int wave = blockIdx.x * 8 + (threadIdx.x >> 5);
    const int lane = threadIdx.x & 31;
    const int row0 = wave << 4;
    if (row0 >= M) return;                    // wave-uniform
    const int m  = lane & 15;                 // A row within tile / B,C column
    const int kh = (lane >> 4) << 1;          // 0 or 2

    const float* __restrict__ xrow = X + (size_t)(row0 + m) * K + kh;
    v2f af[K / 4];
#pragma unroll
    for (int k = 0; k < K; k += 4)
        af[k >> 2] = *(const v2f*)(xrow + k);

#pragma unroll 1
    for (int n0 = 0; n0 < NC; n0 += 16) {
        const float* __restrict__ wp = Wp + (((size_t)(n0 + m)) << 2) + kh;
        v8f c = {0.f, 0.f, 0.f, 0.f, 0.f, 0.f, 0.f, 0.f};
#pragma unroll
        for (int k = 0; k < K; k += 4) {
            const v2f b = *(const v2f*)(wp + (((size_t)(k >> 2) * NC) << 2));
            c = __builtin_amdgcn_wmma_f32_16x16x4_f32(
                    false, af[k >> 2], false, b, (short)0, c, false, false);
        }
        const float bc = bias ? bias[n0 + m] : 0.0f;
        const float we = EXTRA ? wext[n0 + m] : 0.0f;
#pragma unroll
        for (int g = 0; g < 8; ++g) {
            const int row = row0 + g + ((lane >> 4) << 3);
            float val = c[g] + bc;
            if (EXTRA) val += tvec[row] * we;
            if (ACT == 1) val = silu_f(val);
            Y[(size_t)row * NC + (n0 + m)] = val;
        }
    }
}

// ---------------------------------------------------------------------------
// Edge geometry + ExpNormal RBF (padded to 64 cols, cols 50..63 = 0)
// ---------------------------------------------------------------------------
__global__ void edge_geom_kernel(const int* __restrict__ eidx,
                                 const float* __restrict__ pos,
                                 float* __restrict__ cbuf,
                                 float* __restrict__ dhat,
                                 float* __restrict__ fpad)
{
    const int e = blockIdx.x * blockDim.x + threadIdx.x;
    if (e >= NEDGES) return;
    const int i = eidx[e], j = eidx[NEDGES + e];
    const float ex = pos[3 * i]     - pos[3 * j];
    const float ey = pos[3 * i + 1] - pos[3 * j + 1];
    const float ez = pos[3 * i + 2] - pos[3 * j + 2];
    const float r  = sqrtf(ex * ex + ey * ey + ez * ez);
    const float inv = 1.0f / fmaxf(r, 1e-8f);
    dhat[3 * e]     = ex * inv;
    dhat[3 * e + 1] = ey * inv;
    dhat[3 * e + 2] = ez * inv;
    const float picu = 3.14159265358979323846f / CUTF;
    const float cut  = (r < CUTF) ? 0.5f * (__cosf(r * picu) + 1.0f) : 0.0f;
    cbuf[e] = cut;
    const float alpha = 5.0f / CUTF;
    const float start = 4.5399929762484854e-05f;   // exp(-10)
    const float step  = (1.0f - start) / (float)(RBFN - 1);
    const float bb    = (2.0f / (float)RBFN) * (1.0f - start);
    const float beta  = 1.0f / (bb * bb);
    const float ed    = __expf(-alpha * r);
    float* frow = fpad + (size_t)e * KPAD;
    for (int k = 0; k < RBFN; ++k) {
        const float d = ed - (start + (float)k * step);
        frow[k] = cut * __expf(-beta * d * d);
    }
    for (int k = RBFN; k < KPAD; ++k) frow[k] = 0.0f;
}

// ---------------------------------------------------------------------------
// Neighbor-embedding scatter: msg[i] += ne_emb[z[j]] * W[e] * C[e]*(i!=j)
// One wave per edge; 4 channels per lane.
// ---------------------------------------------------------------------------
__global__ __launch_bounds__(256) void ne_scatter_kernel(
    const int* __restrict__ eidx, const int* __restrict__ z,
    const float* __restrict__ ne_emb, const float* __restrict__ Wbuf,
    const float* __restrict__ cbuf, float* __restrict__ msg,
    int e0, int count)
{
    const int er = blockIdx.x * 8 + (threadIdx.x >> 5);
    if (er >= count) return;
    const int e = e0 + er;
    const int i = eidx[e], j = eidx[NEDGES + e];
    if (i == j) return;
    const int lane = threadIdx.x & 31;
    const int c0 = lane * 4;
    const float coef = cbuf[e];
    const float4 w  = *(const float4*)(Wbuf + (size_t)er * HID + c0);
    const float4 em = *(const float4*)(ne_emb + (size_t)z[j] * HID + c0);
    float* dst = msg + (size_t)i * HID + c0;
    unsafeAtomicAdd(dst + 0, em.x * w.x * coef);
    unsafeAtomicAdd(dst + 1, em.y * w.y * coef);
    unsafeAtomicAdd(dst + 2, em.z * w.z * coef);
    unsafeAtomicAdd(dst + 3, em.w * w.w * coef);
}

// Build xcat[n] = [ emb[z[n]] , msg[n] ]
__global__ void xcat_kernel(const int* __restrict__ z,
                            const float* __restrict__ emb,
                            const float* __restrict__ msg,
                            float* __restrict__ xcat)
{
    const int idx = blockIdx.x * blockDim.x + threadIdx.x;
    if (idx >= NNODES * 2 * HID) return;
    const int n = idx >> 8, c = idx & 255;
    xcat[idx] = (c < HID) ? emb[(size_t)z[n] * HID + c]
                          : msg[(size_t)n * HID + (c - HID)];
}

// LayerNorm over 128 features: one wave per row
__global__ __launch_bounds__(256) void ln_kernel(const float* __restrict__ X,
                                                 const float* __restrict__ g,
                                                 const float* __restrict__ b,
                                                 float* __restrict__ Y, int M)
{
    const int row = blockIdx.x * 8 + (threadIdx.x >> 5);
    if (row >= M) return;
    const int lane = threadIdx.x & 31;
    float4 v = ((const float4*)(X + (size_t)row * HID))[lane];
    float s = v.x + v.y + v.z + v.w;
#pragma unroll
    for (int off = 1; off < 32; off <<= 1) s += __shfl_xor(s, off, 32);
    const float mean = s * (1.0f / HID);
    const float d0 = v.x - mean, d1 = v.y - mean, d2 = v.z - mean, d3 = v.w - mean;
    float sq = d0 * d0 + d1 * d1 + d2 * d2 + d3 * d3;
#pragma unroll
    for (int off = 1; off < 32; off <<= 1) sq += __shfl_xor(sq, off, 32);
    const float inv = rsqrtf(sq * (1.0f / HID) + 1e-5f);
    const float4 go = ((const float4*)g)[lane];
    const float4 bo = ((const float4*)b)[lane];
    float4 o;
    o.x = d0 * inv * go.x + bo.x;
    o.y = d1 * inv * go.y + bo.y;
    o.z = d2 * inv * go.z + bo.z;
    o.w = d3 * inv * go.w + bo.w;
    ((float4*)(Y + (size_t)row * HID))[lane] = o;
}

// vec_dot[n,c] = sum_s vp[n,s,c] * vp[n,s,128+c]
__global__ void vecdot_kernel(const float* __restrict__ vp, float* __restrict__ vdot)
{
    const int idx = blockIdx.x * blockDim.x + threadIdx.x;
    if (idx >= NNODES * HID) return;
    const int n = idx >> 7, c = idx & 127;
    const float* base = vp + (size_t)(n * 3) * 384;
    float s = 0.0f;
#pragma unroll
    for (int sp = 0; sp < 3; ++sp)
        s += base[(size_t)sp * 384 + c] * base[(size_t)sp * 384 + HID + c];
    vdot[idx] = s;
}

// ---------------------------------------------------------------------------
// Attention message + scatter. One wave per edge; 4 channels per lane.
// ---------------------------------------------------------------------------
__global__ __launch_bounds__(256) void attn_msg_kernel(
    const int* __restrict__ eidx,
    const float* __restrict__ q,  const float* __restrict__ kk,
    const float* __restrict__ v,  const float* __restrict__ vec,
    const float* __restrict__ dk, const float* __restrict__ dv,
    const float* __restrict__ cbuf, const float* __restrict__ dhat,
    float* __restrict__ xagg, float* __restrict__ vagg, int e0, int count)
{
    const int er = blockIdx.x * 8 + (threadIdx.x >> 5);
    if (er >= count) return;
    const int e = e0 + er;
    const int i = eidx[e], j = eidx[NEDGES + e];
    const int lane = threadIdx.x & 31;
    const int c0 = lane * 4;
    const int h = c0 >> 4;     // head
    const int d = c0 & 15;     // dim-within-head (0,4,8,12)

    const float4 qv  = *(const float4*)(q  + (size_t)i * HID + c0);
    const float4 kv  = *(const float4*)(kk + (size_t)j * HID + c0);
    const float4 dkv = *(const float4*)(dk + (size_t)er * HID + c0);
    float part = qv.x * kv.x * dkv.x + qv.y * kv.y * dkv.y
               + qv.z * kv.z * dkv.z + qv.w * kv.w * dkv.w;
    part += __shfl_xor(part, 1, 32);
    part += __shfl_xor(part, 2, 32);           // per-head sum over 16 dims
    const float attn = silu_f(part) * cbuf[e];

    const float* vb  = v  + (size_t)j  * (3 * HID) + h * 48 + d;
    const float* dvb = dv + (size_t)er * (3 * HID) + h * 48 + d;
    float4 vx = *(const float4*)(vb);
    float4 v1 = *(const float4*)(vb + 16);
    float4 v2 = *(const float4*)(vb + 32);
    const float4 dx_ = *(const float4*)(dvb);
    const float4 d1_ = *(const float4*)(dvb + 16);
    const float4 d2_ = *(const float4*)(dvb + 32);
    vx.x *= dx_.x; vx.y *= dx_.y; vx.z *= dx_.z; vx.w *= dx_.w;
    v1.x *= d1_.x; v1.y *= d1_.y; v1.z *= d1_.z; v1.w *= d1_.w;
    v2.x *= d2_.x; v2.y *= d2_.y; v2.z *= d2_.z; v2.w *= d2_.w;

    float* xr = xagg + (size_t)i * HID + c0;
    unsafeAtomicAdd(xr + 0, vx.x * attn);
    unsafeAtomicAdd(xr + 1, vx.y * attn);
    unsafeAtomicAdd(xr + 2, vx.z * attn);
    unsafeAtomicAdd(xr + 3, vx.w * attn);

    const float dhs[3] = { dhat[3 * e], dhat[3 * e + 1], dhat[3 * e + 2] };
#pragma unroll
    for (int sp = 0; sp < 3; ++sp) {
        const float4 vj = *(const float4*)(vec + (size_t)j * (3 * HID) + sp * HID + c0);
        float* vr = vagg + (size_t)i * (3 * HID) + sp * HID + c0;
        unsafeAtomicAdd(vr + 0, vj.x * v1.x + v2.x * dhs[sp]);
        unsafeAtomicAdd(vr + 1, vj.y * v1.y + v2.y * dhs[sp]);
        unsafeAtomicAdd(vr + 2, vj.z * v1.z + v2.z * dhs[sp]);
        unsafeAtomicAdd(vr + 3, vj.w * v1.w + v2.w * dhs[sp]);
    }
}

// x += vdot*o2 + o3;  vec += vec3*o1 + vagg
__global__ void update_kernel(const float* __restrict__ oo,
                              const float* __restrict__ vdot,
                              const float* __restrict__ vp,
                              const float* __restrict__ vagg,
                              float* __restrict__ x, float* __restrict__ vec)
{
    const int idx = blockIdx.x * blockDim.x + threadIdx.x;
    if (idx >= NNODES * HID) return;
    const int n = idx >> 7, c = idx & 127;
    const float* on = oo + (size_t)n * 384;
    const float o1 = on[c], o2 = on[HID + c], o3 = on[2 * HID + c];
    x[idx] += vdot[idx] * o2 + o3;
    const float* vp3 = vp + (size_t)(n * 3) * 384 + 2 * HID + c;
    float* vn = vec + (size_t)n * (3 * HID) + c;
    const float* va = vagg + (size_t)n * (3 * HID) + c;
#pragma unroll
    for (int sp = 0; sp < 3; ++sp)
        vn[sp * HID] += vp3[(size_t)sp * 384] * o1 + va[sp * HID];
}

// ---------------------------------------------------------------------------
extern "C" void kernel_launch(void* const* d_in, const int* in_sizes, int n_in,
                              void* d_out, int out_size, void* d_ws, size_t ws_size,
                              hipStream_t stream)
{
    (void)in_sizes; (void)n_in; (void)out_size; (void)ws_size;
    const int*   z      = (const int*)d_in[0];
    const float* t      = (const float*)d_in[1];
    const float* pos    = (const float*)d_in[2];
    const int*   eidx   = (const int*)d_in[3];
    const float* emb    = (const float*)d_in[4];
    const float* ne_emb = (const float*)d_in[5];
    const float* ne_dw  = (const float*)d_in[6];
    const float* ne_db  = (const float*)d_in[7];
    const float* ne_cw  = (const float*)d_in[8];
    const float* ne_cb  = (const float*)d_in[9];
    const float* out_g  = (const float*)d_in[10];
    const float* out_b  = (const float*)d_in[11];
    // per-layer params at 12 + l*19:
    // 0 mw1,1 mb1,2 mw2,3 mb2,4 ln_g,5 ln_b,6 qw,7 qb,8 kw,9 kb,
    // 10 vw,11 vb,12 ow,13 ob,14 vecw,15 dkw,16 dkb,17 dvw,18 dvb

    float* ws = (float*)d_ws;
    size_t off = 0;
    auto alloc = [&](size_t nf) { float* p = ws + off; off += (nf + 63) & ~(size_t)63; return p; };
    float* fpad  = alloc((size_t)NEDGES * KPAD);
    float* cbuf  = alloc(NEDGES);
    float* dhat  = alloc((size_t)NEDGES * 3);
    float* dkbuf = alloc((size_t)ECHUNK * HID);
    float* dvbuf = alloc((size_t)ECHUNK * 3 * HID);
    float* msg   = alloc((size_t)NNODES * HID);
    float* xcat  = alloc((size_t)NNODES * 2 * HID);
    float* x     = alloc((size_t)NNODES * HID);
    float* h1    = alloc((size_t)NNODES * HID);
    float* h2    = alloc((size_t)NNODES * HID);
    float* hln   = alloc((size_t)NNODES * HID);
    float* q     = alloc((size_t)NNODES * HID);
    float* kbuf  = alloc((size_t)NNODES * HID);
    float* v     = alloc((size_t)NNODES * 3 * HID);
    float* vp    = alloc((size_t)NNODES * 3 * 3 * HID);
    float* vdot  = alloc((size_t)NNODES * HID);
    float* xagg  = alloc((size_t)NNODES * HID);
    float* vagg  = alloc((size_t)NNODES * 3 * HID);
    float* oo    = alloc((size_t)NNODES * 3 * HID);
    // packed weights
    float* nedw_p = alloc(KPAD * HID);
    float* necw_p = alloc(2 * HID * HID);
    float* dkw_p[NLAYERS];  float* dvw_p[NLAYERS];
    float* mw1_p[NLAYERS];  float* mw2_p[NLAYERS];
    float* qw_p[NLAYERS];   float* kw_p[NLAYERS];
    float* vw_p[NLAYERS];   float* ow_p[NLAYERS];  float* vecw_p[NLAYERS];
    for (int l = 0; l < NLAYERS; ++l) {
        dkw_p[l]  = alloc(KPAD * HID);
        dvw_p[l]  = alloc(KPAD * 3 * HID);
        mw1_p[l]  = alloc(HID * HID);
        mw2_p[l]  = alloc(HID * HID);
        qw_p[l]   = alloc(HID * HID);
        kw_p[l]   = alloc(HID * HID);
        vw_p[l]   = alloc(HID * 3 * HID);
        ow_p[l]   = alloc(HID * 3 * HID);
        vecw_p[l] = alloc(HID * 3 * HID);
    }

    float* xout = (float*)d_out;
    float* vec  = (float*)d_out + (size_t)NNODES * HID;

    auto gblk = [](int M) { return (M / 16 + 7) / 8; };
    auto pblk = [](int n) { return (n + 255) / 256; };
    auto pack = [&](const float* src, float* dst, int krows, int kpad, int nc) {
        pack_weight_kernel<<<pblk(kpad * nc), 256, 0, stream>>>(src, dst, krows, kpad, nc);
    };

    hipMemsetAsync(msg, 0, (size_t)NNODES * HID * 4, stream);
    hipMemsetAsync(vec, 0, (size_t)NNODES * 3 * HID * 4, stream);

    pack(ne_dw, nedw_p, RBFN, KPAD, HID);
    pack(ne_cw, necw_p, 2 * HID, 2 * HID, HID);
    for (int l = 0; l < NLAYERS; ++l) {
        const int pb = 12 + l * 19;
        pack((const float*)d_in[pb + 15], dkw_p[l], RBFN, KPAD, HID);
        pack((const float*)d_in[pb + 17], dvw_p[l], RBFN, KPAD, 3 * HID);
        pack((const float*)d_in[pb + 0],  mw1_p[l], HID, HID, HID);   // first 128 rows
        pack((const float*)d_in[pb + 2],  mw2_p[l], HID, HID, HID);
        pack((const float*)d_in[pb + 6],  qw_p[l],  HID, HID, HID);
        pack((const float*)d_in[pb + 8],  kw_p[l],  HID, HID, HID);
        pack((const float*)d_in[pb + 10], vw_p[l],  HID, HID, 3 * HID);
        pack((const float*)d_in[pb + 12], ow_p[l],  HID, HID, 3 * HID);
        pack((const float*)d_in[pb + 14], vecw_p[l], HID, HID, 3 * HID);
    }

    edge_geom_kernel<<<pblk(NEDGES), 256, 0, stream>>>(eidx, pos, cbuf, dhat, fpad);

    // ---- neighbor embedding ----
    for (int e0 = 0; e0 < NEDGES; e0 += ECHUNK) {
        gemm_wmma<KPAD, HID, 0, false><<<gblk(ECHUNK), 256, 0, stream>>>(
            fpad + (size_t)e0 * KPAD, nedw_p, ne_db, dkbuf, nullptr, nullptr, ECHUNK);
        ne_scatter_kernel<<<(ECHUNK + 7) / 8, 256, 0, stream>>>(
            eidx, z, ne_emb, dkbuf, cbuf, msg, e0, ECHUNK);
    }
    xcat_kernel<<<pblk(NNODES * 2 * HID), 256, 0, stream>>>(z, emb, msg, xcat);
    gemm_wmma<2 * HID, HID, 0, false><<<gblk(NNODES), 256, 0, stream>>>(
        xcat, necw_p, ne_cb, x, nullptr, nullptr, NNODES);

    // ---- attention layers ----
    for (int l = 0; l < NLAYERS; ++l) {
        const int pb = 12 + l * 19;
        const float* mw1   = (const float*)d_in[pb + 0];
        const float* mb1   = (const float*)d_in[pb + 1];
        const float* mb2   = (const float*)d_in[pb + 3];
        const float* ln_g  = (const float*)d_in[pb + 4];
        const float* ln_b  = (const float*)d_in[pb + 5];
        const float* qb    = (const float*)d_in[pb + 7];
        const float* kbias = (const float*)d_in[pb + 9];
        const float* vb    = (const float*)d_in[pb + 11];
        const float* ob    = (const float*)d_in[pb + 13];
        const float* dkb   = (const float*)d_in[pb + 16];
        const float* dvb   = (const float*)d_in[pb + 18];

        // h = LN( silu([x,t]@mw1+mb1) @ mw2 + mb2 )  (row 128 of mw1 as rank-1)
        gemm_wmma<HID, HID, 1, true><<<gblk(NNODES), 256, 0, stream>>>(
            x, mw1_p[l], mb1, h1, t, mw1 + 128 * HID, NNODES);
        gemm_wmma<HID, HID, 0, false><<<gblk(NNODES), 256, 0, stream>>>(
            h1, mw2_p[l], mb2, h2, nullptr, nullptr, NNODES);
        ln_kernel<<<(NNODES + 7) / 8, 256, 0, stream>>>(h2, ln_g, ln_b, hln, NNODES);

        gemm_wmma<HID, HID, 0, false><<<gblk(NNODES), 256, 0, stream>>>(
            hln, qw_p[l], qb, q, nullptr, nullptr, NNODES);
        gemm_wmma<HID, HID, 0, false><<<gblk(NNODES), 256, 0, stream>>>(
            hln, kw_p[l], kbias, kbuf, nullptr, nullptr, NNODES);
        gemm_wmma<HID, 3 * HID, 0, false><<<gblk(NNODES), 256, 0, stream>>>(
            hln, vw_p[l], vb, v, nullptr, nullptr, NNODES);
        gemm_wmma<HID, 3 * HID, 0, false><<<gblk(3 * NNODES), 256, 0, stream>>>(
            vec, vecw_p[l], nullptr, vp, nullptr, nullptr, 3 * NNODES);
        vecdot_kernel<<<pblk(NNODES * HID), 256, 0, stream>>>(vp, vdot);

        hipMemsetAsync(xagg, 0, (size_t)NNODES * HID * 4, stream);
        hipMemsetAsync(vagg, 0, (size_t)NNODES * 3 * HID * 4, stream);

        for (int e0 = 0; e0 < NEDGES; e0 += ECHUNK) {
            gemm_wmma<KPAD, HID, 1, false><<<gblk(ECHUNK), 256, 0, stream>>>(
                fpad + (size_t)e0 * KPAD, dkw_p[l], dkb, dkbuf, nullptr, nullptr, ECHUNK);
            gemm_wmma<KPAD, 3 * HID, 1, false><<<gblk(ECHUNK), 256, 0, stream>>>(
                fpad + (size_t)e0 * KPAD, dvw_p[l], dvb, dvbuf, nullptr, nullptr, ECHUNK);
            attn_msg_kernel<<<(ECHUNK + 7) / 8, 256, 0, stream>>>(
                eidx, q, kbuf, v, vec, dkbuf, dvbuf, cbuf, dhat, xagg, vagg, e0, ECHUNK);
        }

        gemm_wmma<HID, 3 * HID, 0, false><<<gblk(NNODES), 256, 0, stream>>>(
            xagg, ow_p[l], ob, oo, nullptr, nullptr, NNODES);
        update_kernel<<<pblk(NNODES * HID), 256, 0, stream>>>(
            oo, vdot, vp, vagg, x, vec);
    }

    ln_kernel<<<(NNODES + 7) / 8, 256, 0, stream>>>(x, out_g, out_b, xout, NNODES);
}